// Smoothing_Block_12051678232913
// MI455X (gfx1250) — compile-verified
//
#include <hip/hip_runtime.h>

// ---------------------------------------------------------------------------
// Graph smoothing: h <- (h + g*segment_sum(h[src], dst)) * 1/(1+g*D), twice.
// MI455X / gfx1250 strategy:
//   * Working set (~50 MB) fits in the 192 MB L2 -> optimize for L2 gather
//     bandwidth; zero fp atomics (CSR-by-destination built once per launch).
//   * No WMMA: the operator is a 0.03%-dense gather/scatter with no dense
//     K-reduction; matrix pipes cannot be applied without ~1000x wasted work.
//   * CDNA5 paths exercised instead:
//       - global_load_async_to_lds_b32 + s_wait_asynccnt: block-wide async
//         staging of the sortedSrc index window into LDS (frees the VMEM pipe
//         for the random 256B row gathers; indices then read as LDS broadcast)
//       - global_prefetch_b8 (__builtin_prefetch) 8 edges ahead on rows
// ---------------------------------------------------------------------------

constexpr int   DF    = 64;        // feature dim (fixed by reference)
constexpr int   DF4   = DF / 4;    // 16 float4 chunks per row
constexpr float GAMMA = 0.5f;
constexpr int   NODES_PER_BLOCK = 16;   // 256 threads / 16 lanes-per-node
constexpr int   EDGE_CHUNK      = 256;  // staged sortedSrc window (1 KB LDS)

// ---- gfx1250 async Global->LDS helpers (builtin if present, else ISA asm) --
__device__ __forceinline__ void async_g2l_b32(const int* gsrc, int* lds_generic) {
  // Builtin prototype (per compiler diagnostic) takes typed int* operands:
  //   (int addrspace(1)* gsrc, int addrspace(3)* lds, imm offset, imm cpol).
  // AS(3) pointers are 32-bit; the low 32 bits of a flat LDS pointer are the
  // LDS offset (ISA 10.2 aperture rule), so int->ptr casts build the AS views.
#if __has_builtin(__builtin_amdgcn_global_load_async_to_lds_b32)
  typedef __attribute__((address_space(1))) int g_int;
  typedef __attribute__((address_space(3))) int l_int;
  g_int* gp = (g_int*)(unsigned long long)(uintptr_t)gsrc;
  l_int* lp = (l_int*)(unsigned)(uintptr_t)lds_generic;
  __builtin_amdgcn_global_load_async_to_lds_b32(gp, lp, 0, 0);
#else
  unsigned long long ga = (unsigned long long)(uintptr_t)gsrc;
  unsigned           la = (unsigned)(uintptr_t)lds_generic;
  asm volatile("global_load_async_to_lds_b32 %0, %1, off"
               :: "v"(la), "v"(ga) : "memory");
#endif
}

__device__ __forceinline__ void wait_asynccnt0() {
#if __has_builtin(__builtin_amdgcn_s_wait_asynccnt)
  __builtin_amdgcn_s_wait_asynccnt(0);
#else
  asm volatile("s_wait_asynccnt 0" ::: "memory");
#endif
}

// ---------------------------------------------------------------------------

__global__ __launch_bounds__(256) void k_zero_i32(int* __restrict__ p, int n) {
  int i = blockIdx.x * 256 + threadIdx.x;
  if (i < n) p[i] = 0;
}

// counts[d] = in-degree of node d
__global__ __launch_bounds__(256) void k_hist(const int* __restrict__ dst,
                                              int* __restrict__ counts, int nE) {
  int e = blockIdx.x * 256 + threadIdx.x;
  if (e < nE) atomicAdd(&counts[dst[e]], 1);
}

// Single-block exclusive scan (50k elements): per-thread serial chunk sums,
// Hillis-Steele across 1024 partials, then serial write-back.
__global__ __launch_bounds__(1024) void k_scan(const int* __restrict__ counts,
                                               int* __restrict__ offsets, int n) {
  __shared__ int part[1024];
  const int t   = threadIdx.x;
  const int per = (n + 1023) >> 10;
  const int b   = t * per;
  const int e   = (b + per < n) ? (b + per) : n;
  int s = 0;
  for (int i = b; i < e; ++i) s += counts[i];
  part[t] = s;
  __syncthreads();
  for (int d = 1; d < 1024; d <<= 1) {
    int v = (t >= d) ? part[t - d] : 0;
    __syncthreads();
    part[t] += v;
    __syncthreads();
  }
  int run = (t == 0) ? 0 : part[t - 1];   // exclusive base
  for (int i = b; i < e; ++i) { offsets[i] = run; run += counts[i]; }
  if (t == 1023) offsets[n] = run;        // == nE (empty-range threads carry total)
}

// Scatter edges into destination buckets: sortedSrc[offsets[d] ... ] = src ids.
__global__ __launch_bounds__(256) void k_bucket(const int* __restrict__ src,
                                                const int* __restrict__ dst,
                                                const int* __restrict__ offsets,
                                                int* __restrict__ cursor,
                                                int* __restrict__ sortedSrc, int nE) {
  int e = blockIdx.x * 256 + threadIdx.x;
  if (e >= nE) return;
  int d = dst[e];
  int p = offsets[d] + atomicAdd(&cursor[d], 1);
  sortedSrc[p] = src[e];
}

// Fused: hout[i] = (hin[i] + g * sum_{j in in-edges(i)} hin[srcs[j]]) * 1/(1+g*D[i])
// Block = 16 nodes; 16 lanes per node, one float4 chunk per lane.
// The block's contiguous sortedSrc window is staged into LDS via async copy.
__global__ __launch_bounds__(256) void k_agg_update(const float* __restrict__ hin,
                                                    const float* __restrict__ Dv,
                                                    const int* __restrict__ offsets,
                                                    const int* __restrict__ sortedSrc,
                                                    float* __restrict__ hout, int nN) {
  __shared__ int sEdge[EDGE_CHUNK];
  const int t         = threadIdx.x;
  const int nodeBlock = blockIdx.x * NODES_PER_BLOCK;
  const int i         = nodeBlock + (t >> 4);
  const bool valid    = (i < nN);
  const int q4        = (t & 15) << 2;          // feature offset: 0,4,...,60

  const int b = valid ? offsets[i]     : 0;
  const int e = valid ? offsets[i + 1] : 0;

  const int lastNode = (nodeBlock + NODES_PER_BLOCK < nN) ? (nodeBlock + NODES_PER_BLOCK) : nN;
  const int segStart = offsets[nodeBlock];       // block-uniform
  const int segEnd   = offsets[lastNode];        // block-uniform

  float4 acc = make_float4(0.f, 0.f, 0.f, 0.f);

  for (int base = segStart; base < segEnd; base += EDGE_CHUNK) {
    const int cnt = (segEnd - base < EDGE_CHUNK) ? (segEnd - base) : EDGE_CHUNK;
    if (t < cnt) async_g2l_b32(sortedSrc + base + t, &sEdge[t]);  // ASYNCcnt path
    wait_asynccnt0();                                             // own transfers done
    __syncthreads();                                              // all waves done

    const int lo = (b > base) ? b : base;
    const int hi = (e < base + cnt) ? e : (base + cnt);
    for (int j = lo; j < hi; ++j) {
      // gfx1250 prefetch: chunk-0 lane pulls the row 8 edges ahead into cache
      if (((t & 15) == 0) && (j + 8 < hi)) {
        __builtin_prefetch(hin + (size_t)sEdge[j - base + 8] * DF, 0, 3);
      }
      const int s = sEdge[j - base];             // LDS broadcast read
      const float4 v = *reinterpret_cast<const float4*>(hin + (size_t)s * DF + q4);
      acc.x += v.x; acc.y += v.y; acc.z += v.z; acc.w += v.w;
    }
    __syncthreads();                              // before next chunk overwrites sEdge
  }

  if (!valid) return;
  const float  sc = 1.0f / fmaf(GAMMA, Dv[i], 1.0f);
  const float4 hv = *reinterpret_cast<const float4*>(hin + (size_t)i * DF + q4);
  float4 o;
  o.x = fmaf(GAMMA, acc.x, hv.x) * sc;
  o.y = fmaf(GAMMA, acc.y, hv.y) * sc;
  o.z = fmaf(GAMMA, acc.z, hv.z) * sc;
  o.w = fmaf(GAMMA, acc.w, hv.w) * sc;
  *reinterpret_cast<float4*>(hout + (size_t)i * DF + q4) = o;
}

extern "C" void kernel_launch(void* const* d_in, const int* in_sizes, int n_in,
                              void* d_out, int out_size, void* d_ws, size_t ws_size,
                              hipStream_t stream) {
  const float* h0 = (const float*)d_in[0];   // [nN, 64] f32
  const float* Dv = (const float*)d_in[1];   // [nN]     f32
  const int*   ei = (const int*)d_in[2];     // [2, nE]  int (harness int convention)

  const int nN = in_sizes[1];
  const int nE = in_sizes[2] / 2;
  const int* src = ei;
  const int* dst = ei + nE;

  // Workspace layout (all fully rewritten every call):
  //   h1        : nN*64 floats  (iteration-1 result; in-place iter2 would race)
  //   counts    : nN ints
  //   cursor    : nN ints
  //   offsets   : nN+1 ints
  //   sortedSrc : nE ints
  float* h1        = (float*)d_ws;
  int*   counts    = (int*)(h1 + (size_t)nN * DF);
  int*   cursor    = counts + nN;
  int*   offsets   = cursor + nN;
  int*   sortedSrc = offsets + (nN + 1);

  float* out = (float*)d_out;

  const int eBlocks = (nE + 255) / 256;
  const int nBlocks = (nN + NODES_PER_BLOCK - 1) / NODES_PER_BLOCK;
  const int zBlocks = (2 * nN + 255) / 256;

  // Build CSR-by-destination (once; reused by both iterations).
  k_zero_i32 <<<zBlocks, 256, 0, stream>>>(counts, 2 * nN);   // counts + cursor
  k_hist     <<<eBlocks, 256, 0, stream>>>(dst, counts, nE);
  k_scan     <<<1, 1024, 0, stream>>>(counts, offsets, nN);
  k_bucket   <<<eBlocks, 256, 0, stream>>>(src, dst, offsets, cursor, sortedSrc, nE);

  // Two fused smoothing iterations, zero fp atomics.
  k_agg_update<<<nBlocks, 256, 0, stream>>>(h0, Dv, offsets, sortedSrc, h1, nN);
  k_agg_update<<<nBlocks, 256, 0, stream>>>(h1, Dv, offsets, sortedSrc, out, nN);
}